// Signature_4157528342900
// MI455X (gfx1250) — compile-verified
//
#include <hip/hip_runtime.h>

// Depth-5 path signature, B=256 paths, d=10 channels, 127 segments.
// One workgroup per batch. Levels 1..4 + per-step Horner outputs in LDS.
// Level 5 (10000x10) lives in d_out, L2-resident (256*400KB = 102MB < 192MB L2).
// Steps are processed in groups of 8: the per-step rank-1 updates t4_k (x) delta_k
// become a rank-8 update done with two chained v_wmma_f32_16x16x4_f32 per 16x16
// tile, so the level-5 accumulator is loaded/stored once per 8 steps (8x less
// L2 traffic than per-step RMW) and the WMMA K dimension is fully used.

#define DCH   10
#define LEN   128
#define NSEG  127        // LEN-1 segments
#define NB    256
#define SIGSZ 111110     // 10+100+1000+10000+100000
#define OFF5  11110
#define NTIL  625        // 10000 rows / 16
#define NTHR  256
#define NW    8          // waves per workgroup
#define GS    8          // steps per WMMA group (two K=4 WMMAs)
#define NGRP  16         // ceil(127/8)

typedef float v2f __attribute__((ext_vector_type(2)));
typedef float v8f __attribute__((ext_vector_type(8)));

__global__ __launch_bounds__(NTHR)
void sig5_kernel(const float* __restrict__ path, float* __restrict__ out)
{
    const int b    = blockIdx.x;
    const int tid  = threadIdx.x;
    const int lane = tid & 31;
    const int wave = tid >> 5;

    __shared__ float sd[GS * 16];            // delta per step-slot (cols 10..15 stay 0)
    __shared__ float S1[16];
    __shared__ float S2[128];
    __shared__ float S3[1024];
    __shared__ float S4[10016];
    __shared__ float t2[128], x2[128], y2[128];
    __shared__ float t3s[GS * 1024];         // per-slot depth-3 chain for level-5
    __shared__ float x3s[GS * 1024];         // per-slot depth-3 chain for S4

    float*       out5 = out + (size_t)b * SIGSZ + OFF5;
    const float* pb   = path + (size_t)b * DCH * LEN;

    const int n    = lane & 15;              // tile column (N) == this lane's A row (M)
    const int mrow = (lane >> 4) << 3;       // 0 or 8: C-reg row base

    // ---- init: LDS state and level-5 region (tile ownership matches the
    //      update loop, so all global RMW is same-lane/same-address) ----
    for (int i = tid; i < GS * 16; i += NTHR) sd[i] = 0.f;
    for (int i = tid; i < 16;    i += NTHR) S1[i] = 0.f;
    for (int i = tid; i < 128;   i += NTHR) S2[i] = 0.f;
    for (int i = tid; i < 1024;  i += NTHR) S3[i] = 0.f;
    for (int i = tid; i < 10016; i += NTHR) S4[i] = 0.f;
    for (int tile = wave; tile < NTIL; tile += NW) {
        if (n < DCH) {
            float* basep = out5 + (size_t)(tile << 4) * DCH + n;
            #pragma unroll
            for (int j = 0; j < 8; ++j) basep[(mrow + j) * DCH] = 0.f;
        }
    }
    __syncthreads();

    for (int g = 0; g < NGRP; ++g) {
        // ---- per-step chain phases (levels 1..3), fill t3/x3/delta slots ----
        for (int k = 0; k < GS; ++k) {
            const int s = g * GS + k;
            if (tid < DCH) {                       // slot delta (0 for padded steps)
                float d = 0.f;
                if (s < NSEG) { const float* pr = pb + tid * LEN + s; d = pr[1] - pr[0]; }
                sd[k * 16 + tid] = d;
            }
            __syncthreads();
            if (s < NSEG) {                        // uniform branch (s is uniform)
                const float* sdk = sd + k * 16;
                // depth-2 temporaries (depth-1 Horner inlined)
                for (int i = tid; i < 100; i += NTHR) {
                    int p = i / 10, q = i - 10 * p;
                    float dp = sdk[p], dq = sdk[q], s1 = S1[p], s2v = S2[i];
                    t2[i] = s2v + (s1 + dp * 0.2f)        * dq * 0.25f;        // /5,/4
                    x2[i] = s2v + (s1 + dp * 0.25f)       * dq * (1.f / 3.f);  // /4,/3
                    y2[i] = s2v + (s1 + dp * (1.f / 3.f)) * dq * 0.5f;         // /3,/2
                }
                __syncthreads();
                // depth-3 slot fill; S2 update (z-chain inlined)
                for (int i = tid; i < 1000; i += NTHR) {
                    int p = i / 10, q = i - 10 * p;
                    float dq = sdk[q], s3v = S3[i];
                    t3s[k * 1024 + i] = s3v + t2[p] * dq * (1.f / 3.f);
                    x3s[k * 1024 + i] = s3v + x2[p] * dq * 0.5f;
                }
                for (int i = tid; i < 100; i += NTHR) {
                    int p = i / 10, q = i - 10 * p;
                    S2[i] += (S1[p] + sdk[p] * 0.5f) * sdk[q];
                }
                __syncthreads();
                // S3, S1 updates
                for (int i = tid; i < 1000; i += NTHR) {
                    int p = i / 10, q = i - 10 * p;
                    S3[i] += y2[p] * sdk[q];
                }
                if (tid < DCH) S1[tid] += sdk[tid];
                __syncthreads();
            }
        }

        // ---- group-end: rank-8 level-5 accumulate (2x K=4 WMMA per tile),
        //      S4 update fused into the per-row recurrence ----
        {
            v2f bv0, bv1;   // B rows: K0..K3 / K4..K7 = deltas (padded cols are 0)
            bv0.x = (lane < 16) ? sd[0 * 16 + n] : sd[2 * 16 + n];
            bv0.y = (lane < 16) ? sd[1 * 16 + n] : sd[3 * 16 + n];
            bv1.x = (lane < 16) ? sd[4 * 16 + n] : sd[6 * 16 + n];
            bv1.y = (lane < 16) ? sd[5 * 16 + n] : sd[7 * 16 + n];
            for (int tile = wave; tile < NTIL; tile += NW) {
                const int r0 = tile << 4;
                const int r  = r0 + n;               // this lane's level-4 row
                const int p  = r / 10, q = r - 10 * p;
                float run = S4[r];                   // S4 frozen at group start
                float u[GS];
                #pragma unroll
                for (int k = 0; k < GS; ++k) {
                    float dq = sd[k * 16 + q];
                    u[k]  = run + t3s[k * 1024 + p] * dq * 0.5f;   // t4_k row value
                    run  += x3s[k * 1024 + p] * dq;                // advance S4
                }
                if (lane < 16) S4[r] = run;          // fused S4 update (wave-private)
                v2f av0, av1;                        // A cols: K0..K3 / K4..K7 = u_k
                av0.x = (lane < 16) ? u[0] : u[2];
                av0.y = (lane < 16) ? u[1] : u[3];
                av1.x = (lane < 16) ? u[4] : u[6];
                av1.y = (lane < 16) ? u[5] : u[7];
                v8f cv;
                float* basep = out5 + (size_t)r0 * DCH + n;
                if (n < DCH) {
                    #pragma unroll
                    for (int j = 0; j < 8; ++j) cv[j] = basep[(mrow + j) * DCH];
                } else {
                    #pragma unroll
                    for (int j = 0; j < 8; ++j) cv[j] = 0.f;
                }
                cv = __builtin_amdgcn_wmma_f32_16x16x4_f32(
                        false, av0, false, bv0, (short)0, cv, false, false);
                cv = __builtin_amdgcn_wmma_f32_16x16x4_f32(
                        false, av1, false, bv1, (short)0, cv, false, false);
                if (n < DCH) {
                    #pragma unroll
                    for (int j = 0; j < 8; ++j) basep[(mrow + j) * DCH] = cv[j];
                }
            }
        }
        __syncthreads();   // protect t3s/x3s/sd reuse by next group
    }

    // ---- epilogue: levels 1..4 (level 5 already accumulated in place) ----
    float* ob = out + (size_t)b * SIGSZ;
    if (tid < DCH) ob[tid] = S1[tid];
    for (int i = tid; i < 100;   i += NTHR) ob[10 + i]   = S2[i];
    for (int i = tid; i < 1000;  i += NTHR) ob[110 + i]  = S3[i];
    for (int i = tid; i < 10000; i += NTHR) ob[1110 + i] = S4[i];
}

extern "C" void kernel_launch(void* const* d_in, const int* in_sizes, int n_in,
                              void* d_out, int out_size, void* d_ws, size_t ws_size,
                              hipStream_t stream) {
    (void)in_sizes; (void)n_in; (void)d_ws; (void)ws_size; (void)out_size;
    const float* path = (const float*)d_in[0];   // (256, 10, 128) f32
    float*       out  = (float*)d_out;           // (256, 111110) f32
    sig5_kernel<<<NB, NTHR, 0, stream>>>(path, out);
}